// CrossEntropyWithPerClassFocal_24000277250099
// MI455X (gfx1250) — compile-verified
//
#include <hip/hip_runtime.h>

#define NCLS 19
#define HW_SHIFT 18            // 512*512 = 2^18
#define HW (1 << HW_SHIFT)
#define IGN 255
#define EPS 1e-6f

typedef float v8f __attribute__((ext_vector_type(8)));
typedef float v2f __attribute__((ext_vector_type(2)));

// ws layout: [0]=ce_sum  [1]=n_valid  [2..20]=class focal sums  [21..39]=class counts
__global__ void ce_focal_init(float* ws) {
    int i = threadIdx.x;
    if (i < 40) ws[i] = 0.0f;
}

__global__ __launch_bounds__(256) void ce_focal_main(
    const float* __restrict__ logits, const long long* __restrict__ tgt,
    float* __restrict__ ws, int n_groups)
{
    __shared__ float ls_sum[NCLS];
    __shared__ float ls_cnt[NCLS];
    __shared__ float ls_ce;
    __shared__ float ls_nv;
    const int tid = threadIdx.x;
    if (tid < NCLS) { ls_sum[tid] = 0.f; ls_cnt[tid] = 0.f; }
    if (tid == NCLS)     ls_ce = 0.f;
    if (tid == NCLS + 1) ls_nv = 0.f;
    __syncthreads();

    float ce_acc = 0.f, nv_acc = 0.f;
    long g = (long)blockIdx.x * blockDim.x + tid;   // one 4-pixel group per thread
    if (g < (long)n_groups) {
        long p  = g << 2;
        int  b  = (int)(p >> HW_SHIFT);
        int  hw = (int)(p & (HW - 1));
        const float* base = logits + (size_t)b * NCLS * HW + hw;

        // 19 coalesced b128 loads: one float4 (4 pixels) per channel plane
        float xr[NCLS][4];
        #pragma unroll
        for (int c = 0; c < NCLS; ++c) {
            float4 v = *reinterpret_cast<const float4*>(base + (size_t)c * HW);
            xr[c][0] = v.x; xr[c][1] = v.y; xr[c][2] = v.z; xr[c][3] = v.w;
        }
        long long t4[4];
        #pragma unroll
        for (int i = 0; i < 4; ++i) t4[i] = tgt[p + i];

        #pragma unroll
        for (int i = 0; i < 4; ++i) {
            long long t = t4[i];
            bool valid = (t != IGN);
            int tc = (int)t;
            tc = tc < 0 ? 0 : (tc > NCLS - 1 ? NCLS - 1 : tc);

            float m = xr[0][i], xt = xr[0][i];
            #pragma unroll
            for (int c = 1; c < NCLS; ++c) {     // max + pick target logit (static idx)
                float v = xr[c][i];
                m  = fmaxf(m, v);
                xt = (c == tc) ? v : xt;
            }
            float s = 0.f;
            #pragma unroll
            for (int c = 0; c < NCLS; ++c) s += __expf(xr[c][i] - m);
            float lse = m + __logf(s);
            float nll = lse - xt;
            if (valid) {
                ce_acc += nll; nv_acc += 1.f;
                float pt = __expf(-nll);
                pt = fminf(fmaxf(pt, EPS), 1.f);
                float om = 1.f - pt;
                float fl = -__logf(pt) * om * om * om;
                atomicAdd(&ls_sum[tc], fl);      // ds_add_f32
                atomicAdd(&ls_cnt[tc], 1.f);
            }
        }
    }

    // wave32 reduction of ce / n_valid
    #pragma unroll
    for (int off = 16; off > 0; off >>= 1) {
        ce_acc += __shfl_down(ce_acc, off);
        nv_acc += __shfl_down(nv_acc, off);
    }
    if ((tid & 31) == 0) {
        atomicAdd(&ls_ce, ce_acc);
        atomicAdd(&ls_nv, nv_acc);
    }
    __syncthreads();

    if (tid < NCLS) {
        atomicAdd(&ws[2  + tid], ls_sum[tid]);
        atomicAdd(&ws[21 + tid], ls_cnt[tid]);
    }
    if (tid == 32) {
        atomicAdd(&ws[0], ls_ce);
        atomicAdd(&ws[1], ls_nv);
    }
}

// One wave. Exact f32 WMMA (16x16x4) used as a cross-lane reducer:
//   A(m,0)=mean[m], A(m,1)=mean[16+m] (m<3), K=2/3 zero; B = all ones.
//   D(m,n) = sum_k A(m,k)  -> row sums; then sum rows via column 0
//   (lane 0 holds rows 0-7 in D[0..7], lane 16 holds rows 8-15).
__global__ __launch_bounds__(32) void ce_focal_final(
    const float* __restrict__ ws, float* __restrict__ out)
{
    const int L = threadIdx.x;
    const int m = L & 15;
    bool lo = (L < 16);

    float s0  = ws[2  + m];
    float n0c = ws[21 + m];
    float s1  = (m < 3) ? ws[2  + 16 + m] : 0.f;
    float n1c = (m < 3) ? ws[21 + 16 + m] : 0.f;

    float mean0 = lo ? s0 / fmaxf(n0c, 1.f) : 0.f;
    float mean1 = lo ? s1 / fmaxf(n1c, 1.f) : 0.f;
    float pres0 = lo ? (n0c > 0.f ? 1.f : 0.f) : 0.f;
    float pres1 = lo ? (n1c > 0.f ? 1.f : 0.f) : 0.f;

    v2f A1; A1[0] = mean0; A1[1] = mean1;
    v2f A2; A2[0] = pres0; A2[1] = pres1;
    v2f B1; B1[0] = 1.f;   B1[1] = 1.f;
    v8f D1 = {0.f,0.f,0.f,0.f,0.f,0.f,0.f,0.f};
    v8f D2 = {0.f,0.f,0.f,0.f,0.f,0.f,0.f,0.f};

    D1 = __builtin_amdgcn_wmma_f32_16x16x4_f32(false, A1, false, B1, (short)0, D1, false, false);
    D2 = __builtin_amdgcn_wmma_f32_16x16x4_f32(false, A2, false, B1, (short)0, D2, false, false);

    float r1 = 0.f, r2 = 0.f;
    #pragma unroll
    for (int j = 0; j < 8; ++j) { r1 += D1[j]; r2 += D2[j]; }

    float sum_means = __shfl(r1, 0) + __shfl(r1, 16);
    float n_present = __shfl(r2, 0) + __shfl(r2, 16);

    if (L == 0) {
        float ce    = ws[0] / fmaxf(ws[1], 1.f);
        float focal = sum_means / fmaxf(n_present, 1.f);
        out[0] = ce + 1.0f * focal;   // FOCAL_WEIGHT = 1.0
    }
}

extern "C" void kernel_launch(void* const* d_in, const int* in_sizes, int n_in,
                              void* d_out, int out_size, void* d_ws, size_t ws_size,
                              hipStream_t stream)
{
    const float*     logits = (const float*)d_in[0];
    const long long* tgt    = (const long long*)d_in[1];
    float* ws  = (float*)d_ws;
    float* out = (float*)d_out;

    int npix     = in_sizes[1];      // 8*512*512
    int n_groups = npix >> 2;        // 4 pixels per thread
    int blocks   = (n_groups + 255) / 256;

    hipLaunchKernelGGL(ce_focal_init,  dim3(1),      dim3(64), 0, stream, ws);
    hipLaunchKernelGGL(ce_focal_main,  dim3(blocks), dim3(256),0, stream, logits, tgt, ws, n_groups);
    hipLaunchKernelGGL(ce_focal_final, dim3(1),      dim3(32), 0, stream, ws, out);
}